// SSIM_34144990003811
// MI455X (gfx1250) — compile-verified
//
#include <hip/hip_runtime.h>

typedef float v2f __attribute__((ext_vector_type(2)));
typedef float v8f __attribute__((ext_vector_type(8)));

#define IMG        512
#define OUTW       502
#define NTILE      32                    // 16-wide tiles per dim (ceil(502/16))
#define NPLANES    96                    // 32 batch * 3 channels
#define TILES_TOT  (NPLANES * NTILE * NTILE)   // 98304
#define WPB        4                     // waves per block (wave32)
#define LDS_ROW    20                    // 16 cols + 4 pad -> halfwaves hit disjoint banks
#define LDS_SIG    (32 * LDS_ROW)        // 640 floats per signal (32 rows staged)
#define LDS_WAVE   (5 * LDS_SIG)         // 3200 floats per wave

// D = A(16x4) * B(4x16) + C(16x16), all fp32, full-precision MACs on the matrix pipe.
__device__ __forceinline__ v8f wmma4(v2f a, v2f b, v8f c) {
    return __builtin_amdgcn_wmma_f32_16x16x4_f32(
        /*neg_a=*/false, a, /*neg_b=*/false, b,
        /*c_mod=*/(short)0, c, /*reuse_a=*/false, /*reuse_b=*/false);
}

__global__ void ssim_zero(double* acc) { acc[0] = 0.0; }

__global__ __launch_bounds__(128)
void ssim_main(const float* __restrict__ X, const float* __restrict__ Y,
               double* __restrict__ acc)
{
    __shared__ float lds[WPB * LDS_WAVE];          // 51200 B

    const int tid  = threadIdx.x;
    const int wave = tid >> 5;
    const int lane = tid & 31;
    const int c    = lane & 15;                    // column / M-row within tile
    const int h    = lane >> 4;                    // halfwave
    float* W = &lds[wave * LDS_WAVE];

    // ---- Gaussian taps g[0..10], sigma=1.5, normalized (matches reference) ----
    float gsum = 0.f;
    #pragma unroll
    for (int t = 0; t < 11; ++t) {
        float d = (float)(t - 5);
        gsum += __expf(-d * d / 4.5f);
    }
    const float ginv = 1.f / gsum;

    // Band registers: band[kc] = { w(kc*4+2h   - c),
    //                              w(kc*4+2h+1 - c) },  w(d)=g[d] if 0<=d<=10 else 0.
    // Serves as the horizontal B operand AND the vertical A operand (both are g[j - lane%16]).
    v2f band[7];
    #pragma unroll
    for (int kc = 0; kc < 7; ++kc) {
        int j0 = kc * 4 + 2 * h;
        int d0 = j0 - c, d1 = j0 + 1 - c;
        float f0 = (float)(d0 - 5), f1 = (float)(d1 - 5);
        band[kc].x = (d0 >= 0 && d0 <= 10) ? __expf(-f0 * f0 / 4.5f) * ginv : 0.f;
        band[kc].y = (d1 >= 0 && d1 <= 10) ? __expf(-f1 * f1 / 4.5f) * ginv : 0.f;
    }

    // ---- Tile decode: one 16x16 output tile per wave ----
    const int tileId = blockIdx.x * WPB + wave;
    const int img = tileId >> 10;                  // / (32*32)
    const int rem = tileId & 1023;
    const int ty  = rem >> 5;
    const int tx  = rem & 31;
    const long base = (long)img * (IMG * IMG);
    const int r0 = ty * 16, c0 = tx * 16;

    // ---- Horizontal pass: H[26x16] per signal via 7 banded K=4 WMMAs x 2 row-tiles ----
    v8f Hacc[5][2];
    #pragma unroll
    for (int s = 0; s < 5; ++s) {
        Hacc[s][0] = (v8f){0,0,0,0,0,0,0,0};
        Hacc[s][1] = (v8f){0,0,0,0,0,0,0,0};
    }

    #pragma unroll
    for (int t = 0; t < 2; ++t) {
        int row = r0 + t * 16 + c;                 // A-matrix M row = lane%16
        if (row > IMG - 1) row = IMG - 1;          // clamp: only feeds masked outputs
        const float* xr = X + base + (long)row * IMG;
        const float* yr = Y + base + (long)row * IMG;
        #pragma unroll
        for (int kc = 0; kc < 7; ++kc) {
            int col0 = c0 + kc * 4 + 2 * h;        // K = kc*4 + r + 2h, r in {0,1}
            int col1 = col0 + 1;
            if (col0 > IMG - 1) col0 = IMG - 1;
            if (col1 > IMG - 1) col1 = IMG - 1;
            float x0 = (xr[col0] + 1.f) * 0.5f, x1 = (xr[col1] + 1.f) * 0.5f;
            float y0 = (yr[col0] + 1.f) * 0.5f, y1 = (yr[col1] + 1.f) * 0.5f;
            v2f ax  = {x0,      x1     };
            v2f ay  = {y0,      y1     };
            v2f axx = {x0 * x0, x1 * x1};
            v2f ayy = {y0 * y0, y1 * y1};
            v2f axy = {x0 * y0, x1 * y1};
            Hacc[0][t] = wmma4(ax,  band[kc], Hacc[0][t]);
            Hacc[1][t] = wmma4(ay,  band[kc], Hacc[1][t]);
            Hacc[2][t] = wmma4(axx, band[kc], Hacc[2][t]);
            Hacc[3][t] = wmma4(ayy, band[kc], Hacc[3][t]);
            Hacc[4][t] = wmma4(axy, band[kc], Hacc[4][t]);
        }
    }

    // ---- Stage intermediate to LDS in plain row-major (D layout: row = r + 8h) ----
    #pragma unroll
    for (int s = 0; s < 5; ++s)
        #pragma unroll
        for (int t = 0; t < 2; ++t)
            #pragma unroll
            for (int r = 0; r < 8; ++r)
                W[s * LDS_SIG + (t * 16 + r + 8 * h) * LDS_ROW + c] = Hacc[s][t][r];

    __syncthreads();

    // ---- Vertical pass: out[16x16] = Gv(band) x H, 7 K=4 WMMAs per signal ----
    v8f O[5];
    #pragma unroll
    for (int s = 0; s < 5; ++s) O[s] = (v8f){0,0,0,0,0,0,0,0};

    #pragma unroll
    for (int kc = 0; kc < 7; ++kc) {
        int j0 = kc * 4 + 2 * h;                   // B layout: K = reg + 2h
        #pragma unroll
        for (int s = 0; s < 5; ++s) {
            v2f b;
            b.x = W[s * LDS_SIG + j0       * LDS_ROW + c];
            b.y = W[s * LDS_SIG + (j0 + 1) * LDS_ROW + c];
            O[s] = wmma4(band[kc], b, O[s]);
        }
    }

    // ---- Elementwise SSIM map + masked mean ----
    const float C1 = 0.0001f;                      // (0.01*1)^2
    const float C2 = 0.0009f;                      // (0.03*1)^2
    float sum = 0.f;
    #pragma unroll
    for (int r = 0; r < 8; ++r) {
        int orow = r0 + r + 8 * h;                 // D layout row
        int ocol = c0 + c;
        bool valid = (orow < OUTW) && (ocol < OUTW);
        float mu1 = O[0][r], mu2 = O[1][r];
        float exx = O[2][r], eyy = O[3][r], exy = O[4][r];
        float m11 = mu1 * mu1, m22 = mu2 * mu2, m12 = mu1 * mu2;
        float s1 = exx - m11, s2 = eyy - m22, s12 = exy - m12;
        float cs  = (2.f * s12 + C2) / (s1 + s2 + C2);
        float lum = (2.f * m12 + C1) / (m11 + m22 + C1);
        sum += valid ? lum * cs : 0.f;
    }

    // wave32 reduction, then one f64 atomic per wave
    #pragma unroll
    for (int off = 16; off > 0; off >>= 1)
        sum += __shfl_xor(sum, off, 32);
    if (lane == 0)
        atomicAdd(acc, (double)sum);
}

__global__ void ssim_final(const double* __restrict__ acc, float* __restrict__ out) {
    out[0] = (float)(acc[0] * (1.0 / 24192384.0));   // 32*3*502*502
}

extern "C" void kernel_launch(void* const* d_in, const int* in_sizes, int n_in,
                              void* d_out, int out_size, void* d_ws, size_t ws_size,
                              hipStream_t stream) {
    (void)in_sizes; (void)n_in; (void)out_size; (void)ws_size;
    const float* X = (const float*)d_in[0];
    const float* Y = (const float*)d_in[1];
    double* acc = (double*)d_ws;

    ssim_zero <<<1, 1, 0, stream>>>(acc);
    ssim_main <<<TILES_TOT / WPB, 128, 0, stream>>>(X, Y, acc);
    ssim_final<<<1, 1, 0, stream>>>(acc, (float*)d_out);
}